// GNNModel_12446815224362
// MI455X (gfx1250) — compile-verified
//
#include <hip/hip_runtime.h>
#include <hip/hip_bf16.h>

typedef float v2f __attribute__((ext_vector_type(2)));
typedef float v8f __attribute__((ext_vector_type(8)));

#define DIM 128
#define WPITCH 68        // LDS pitch for transposed W: bank stride 4 -> conflict-free
#define SCAN_BLK 1024

// ---------------------------------------------------------------------------
// In-degree histogram (int), one thread per edge/self-loop
// ---------------------------------------------------------------------------
__global__ void count_kernel(const int* __restrict__ dst, int* __restrict__ cnt,
                             int nE, int n) {
    long long i = (long long)blockIdx.x * blockDim.x + threadIdx.x;
    if (i < nE) {
        atomicAdd(&cnt[dst[i]], 1);
    } else if (i < (long long)nE + n) {
        atomicAdd(&cnt[i - nE], 1);          // self-loop
    }
}

// dis[i] = deg>0 ? rsqrt(max(deg,1)) : 0   (deg = cnt, incl. self-loop)
__global__ void dis_kernel(const int* __restrict__ cnt, float* __restrict__ dis, int n) {
    int i = blockIdx.x * blockDim.x + threadIdx.x;
    if (i < n) {
        float d = (float)cnt[i];
        dis[i] = d > 0.0f ? rsqrtf(fmaxf(d, 1.0f)) : 0.0f;
    }
}

// ---------------------------------------------------------------------------
// Exclusive scan (3-kernel): per-block Hillis-Steele, scan of block sums, add
// ---------------------------------------------------------------------------
__global__ void __launch_bounds__(SCAN_BLK)
scan_block_kernel(const int* __restrict__ in, int* __restrict__ out,
                  int* __restrict__ bsums, int n) {
    __shared__ int sm[SCAN_BLK];
    int i = blockIdx.x * SCAN_BLK + threadIdx.x;
    int v = (i < n) ? in[i] : 0;
    sm[threadIdx.x] = v;
    __syncthreads();
    for (int off = 1; off < SCAN_BLK; off <<= 1) {
        int t = (threadIdx.x >= off) ? sm[threadIdx.x - off] : 0;
        __syncthreads();
        sm[threadIdx.x] += t;
        __syncthreads();
    }
    if (i < n) out[i] = sm[threadIdx.x] - v;               // exclusive
    if (threadIdx.x == SCAN_BLK - 1) bsums[blockIdx.x] = sm[SCAN_BLK - 1];
}

__global__ void __launch_bounds__(SCAN_BLK)
scan_tops_kernel(int* __restrict__ bsums, int nb) {        // nb <= 1024, one block
    __shared__ int sm[SCAN_BLK];
    int v = (threadIdx.x < nb) ? bsums[threadIdx.x] : 0;
    sm[threadIdx.x] = v;
    __syncthreads();
    for (int off = 1; off < SCAN_BLK; off <<= 1) {
        int t = (threadIdx.x >= off) ? sm[threadIdx.x - off] : 0;
        __syncthreads();
        sm[threadIdx.x] += t;
        __syncthreads();
    }
    if (threadIdx.x < nb) bsums[threadIdx.x] = sm[threadIdx.x] - v;  // exclusive
}

__global__ void scan_add_kernel(int* __restrict__ out, const int* __restrict__ bsums,
                                int n, int total) {
    int i = blockIdx.x * blockDim.x + threadIdx.x;
    if (i < n) out[i] += bsums[i >> 10];
    if (i == 0) out[n] = total;
}

__global__ void copy_int_kernel(const int* __restrict__ in, int* __restrict__ out, int n) {
    int i = blockIdx.x * blockDim.x + threadIdx.x;
    if (i < n) out[i] = in[i];
}

// CSR bucket fill: csr_src[pos] = src for each edge (incl. self-loops)
__global__ void fill_kernel(const int* __restrict__ src, const int* __restrict__ dst,
                            int* __restrict__ cursor, int* __restrict__ csr_src,
                            int nE, int n) {
    long long i = (long long)blockIdx.x * blockDim.x + threadIdx.x;
    int s, d;
    if (i < nE)                    { s = src[i]; d = dst[i]; }
    else if (i < (long long)nE + n){ s = d = (int)(i - nE); }
    else return;
    int pos = atomicAdd(&cursor[d], 1);
    csr_src[pos] = s;
}

// ---------------------------------------------------------------------------
// WMMA GEMM: Y[n,128] = X[n,128] @ W[128,128]   (fp32, V_WMMA_F32_16X16X4_F32)
// 8 waves/block, each wave computes a 16-row x 128-col strip.
// W^T staged in LDS with pitch 68 (conflict-free ds_load_b64 B fragments).
// ---------------------------------------------------------------------------
__global__ void __launch_bounds__(256)
gemm_wmma_f32(const float* __restrict__ X, const float* __restrict__ W,
              float* __restrict__ Y, int n) {
    __shared__ float Wl[DIM * WPITCH];                     // 34 KB
    // transposed cooperative load: Wl[n*68 + k] = W[k*128 + n]
    for (int idx = threadIdx.x; idx < DIM * DIM; idx += 256) {
        int k = idx >> 7, c = idx & 127;
        Wl[c * WPITCH + k] = W[idx];
    }
    __syncthreads();

    int tid  = threadIdx.x;
    int wave = tid >> 5;
    int lane = tid & 31;
    int rowBase = (blockIdx.x * 8 + wave) * 16;

    // A frag (16x4 f32): lane<16 -> M=lane, K=k0+{0,1}; lane>=16 -> M=lane-16, K=k0+{2,3}
    int m    = lane & 15;
    int koff = (lane >> 4) * 2;
    int row  = rowBase + m;
    int rowc = row < n ? row : (n - 1);                    // clamp loads, guard stores
    const float* xrow = X + (size_t)rowc * DIM;

    v8f acc[8];
#pragma unroll
    for (int j = 0; j < 8; ++j) acc[j] = {};

    for (int k0 = 0; k0 < DIM; k0 += 4) {
        v2f a = *(const v2f*)(xrow + k0 + koff);
#pragma unroll
        for (int j = 0; j < 8; ++j) {
            // B frag (4x16): lane<16 -> N=lane, K=k0+{0,1}; lane>=16 -> N=lane-16, K=k0+{2,3}
            v2f b = *(const v2f*)&Wl[(j * 16 + m) * WPITCH + k0 + koff];
            acc[j] = __builtin_amdgcn_wmma_f32_16x16x4_f32(
                false, a, false, b, (short)0, acc[j], false, false);
        }
    }

    // C/D: VGPR r, lanes 0-15 -> M=r, lanes 16-31 -> M=r+8; N = lane&15
    int outRow = rowBase + (lane >> 4) * 8;
    int ncol   = lane & 15;
#pragma unroll
    for (int j = 0; j < 8; ++j) {
#pragma unroll
        for (int r = 0; r < 8; ++r) {
            int rr = outRow + r;
            if (rr < n) Y[(size_t)rr * DIM + j * 16 + ncol] = acc[j][r];
        }
    }
}

// ---------------------------------------------------------------------------
// Propagate by gather (no atomics): one wave per destination node.
// out[d] = relu( sum_{e in CSR[d]} h[src_e] * dis[src_e]*dis[d]  + bias )
// ---------------------------------------------------------------------------
__global__ void gather_kernel(const float* __restrict__ H,
                              const int* __restrict__ csr_src,
                              const int* __restrict__ offsets,
                              const float* __restrict__ dis,
                              const float* __restrict__ bias,
                              float* __restrict__ out, int n) {
    int node = blockIdx.x * (blockDim.x >> 5) + (threadIdx.x >> 5);
    int lane = threadIdx.x & 31;
    if (node >= n) return;
    int beg = offsets[node];
    int end = offsets[node + 1];
    float dd = dis[node];
    float4 acc = {0.f, 0.f, 0.f, 0.f};
#pragma unroll 4
    for (int j = beg; j < end; ++j) {
        int s = csr_src[j];                                // wave-uniform -> s_load
        float w = dis[s] * dd;
        float4 v = ((const float4*)(H + (size_t)s * DIM))[lane];
        acc.x += v.x * w; acc.y += v.y * w;
        acc.z += v.z * w; acc.w += v.w * w;
    }
    float4 b = ((const float4*)bias)[lane];
    acc.x = fmaxf(acc.x + b.x, 0.f);
    acc.y = fmaxf(acc.y + b.y, 0.f);
    acc.z = fmaxf(acc.z + b.z, 0.f);
    acc.w = fmaxf(acc.w + b.w, 0.f);
    ((float4*)(out + (size_t)node * DIM))[lane] = acc;
}

// ---------------------------------------------------------------------------
// Mean-pool accumulation: one wave per node (fp atomics, small volume)
// ---------------------------------------------------------------------------
__global__ void pool_kernel(const float* __restrict__ h, const int* __restrict__ batch,
                            float* __restrict__ sums, float* __restrict__ cnts, int n) {
    int node = blockIdx.x * (blockDim.x >> 5) + (threadIdx.x >> 5);
    int lane = threadIdx.x & 31;
    if (node >= n) return;
    int g = batch[node];
    float4 v = ((const float4*)(h + (size_t)node * DIM))[lane];
    float* sp = sums + (size_t)g * DIM + lane * 4;
    atomicAdd(sp + 0, v.x);
    atomicAdd(sp + 1, v.y);
    atomicAdd(sp + 2, v.z);
    atomicAdd(sp + 3, v.w);
    if (lane == 0) atomicAdd(&cnts[g], 1.0f);
}

// out[g] = (sums[g]/max(cnt,1)) . Wfc + bfc   — one wave per graph
__global__ void final_kernel(const float* __restrict__ sums, const float* __restrict__ cnts,
                             const float* __restrict__ Wfc, const float* __restrict__ bfc,
                             float* __restrict__ out, int G) {
    int g    = blockIdx.x * (blockDim.x >> 5) + (threadIdx.x >> 5);
    int lane = threadIdx.x & 31;
    if (g >= G) return;
    float inv = 1.0f / fmaxf(cnts[g], 1.0f);
    float4 s = ((const float4*)(sums + (size_t)g * DIM))[lane];
    float4 w = ((const float4*)Wfc)[lane];
    float acc = (s.x * w.x + s.y * w.y + s.z * w.z + s.w * w.w) * inv;
#pragma unroll
    for (int o = 16; o > 0; o >>= 1) acc += __shfl_down(acc, o, 32);
    if (lane == 0) out[g] = acc + bfc[0];
}

// ---------------------------------------------------------------------------
extern "C" void kernel_launch(void* const* d_in, const int* in_sizes, int n_in,
                              void* d_out, int out_size, void* d_ws, size_t ws_size,
                              hipStream_t stream) {
    const float* x   = (const float*)d_in[0];
    const int*   ei  = (const int*)d_in[1];
    const int*   bat = (const int*)d_in[2];
    const float* W1  = (const float*)d_in[4];
    const float* b1  = (const float*)d_in[5];
    const float* W2  = (const float*)d_in[6];
    const float* b2  = (const float*)d_in[7];
    const float* Wfc = (const float*)d_in[8];
    const float* bfc = (const float*)d_in[9];
    float*       out = (float*)d_out;

    const int N = in_sizes[0] / DIM;
    const int E = in_sizes[1] / 2;
    const int G = out_size;
    const int* src = ei;               // edge_index[0]
    const int* dst = ei + E;           // edge_index[1]
    const int total = E + N;           // edges + self-loops

    // workspace layout (all 4-byte types; float4 users first, 16B-aligned)
    float* bufA    = (float*)d_ws;                       // N*128
    float* bufB    = bufA + (size_t)N * DIM;             // N*128
    float* sums    = bufB + (size_t)N * DIM;             // G*128
    float* dis     = sums + (size_t)G * DIM;             // N
    float* cnts    = dis + N;                            // G
    int*   offsets = (int*)(cnts + G);                   // N+1
    int*   cntcur  = offsets + (N + 1);                  // N (counts, then cursor)
    int*   bsums   = cntcur + N;                         // <=1024
    int*   csr_src = bsums + SCAN_BLK;                   // E+N

    const int scanBlocks = (N + SCAN_BLK - 1) / SCAN_BLK;      // <= 1024 for N <= 1M
    const int edgeBlocks = (total + 255) / 256;
    const int nodeBlocks = (N + 255) / 256;
    const int waveNodeBlocks = (N * 32 + 255) / 256;           // one wave per node
    const int gemmBlocks = (N + 127) / 128;

    // --- CSR build + normalization ----------------------------------------
    hipMemsetAsync(cntcur, 0, (size_t)N * sizeof(int), stream);
    count_kernel<<<edgeBlocks, 256, 0, stream>>>(dst, cntcur, E, N);
    dis_kernel<<<nodeBlocks, 256, 0, stream>>>(cntcur, dis, N);
    scan_block_kernel<<<scanBlocks, SCAN_BLK, 0, stream>>>(cntcur, offsets, bsums, N);
    scan_tops_kernel<<<1, SCAN_BLK, 0, stream>>>(bsums, scanBlocks);
    scan_add_kernel<<<nodeBlocks, 256, 0, stream>>>(offsets, bsums, N, total);
    copy_int_kernel<<<nodeBlocks, 256, 0, stream>>>(offsets, cntcur, N);
    fill_kernel<<<edgeBlocks, 256, 0, stream>>>(src, dst, cntcur, csr_src, E, N);

    // --- layer 1: bufA = x@W1 ; bufB = relu(prop(bufA) + b1) ---------------
    gemm_wmma_f32<<<gemmBlocks, 256, 0, stream>>>(x, W1, bufA, N);
    gather_kernel<<<waveNodeBlocks, 256, 0, stream>>>(bufA, csr_src, offsets, dis, b1, bufB, N);

    // --- layer 2: bufA = bufB@W2 ; bufB = relu(prop(bufA) + b2) ------------
    gemm_wmma_f32<<<gemmBlocks, 256, 0, stream>>>(bufB, W2, bufA, N);
    gather_kernel<<<waveNodeBlocks, 256, 0, stream>>>(bufA, csr_src, offsets, dis, b2, bufB, N);

    // --- pooling + head ----------------------------------------------------
    hipMemsetAsync(sums, 0, (size_t)G * DIM * sizeof(float), stream);
    hipMemsetAsync(cnts, 0, (size_t)G * sizeof(float), stream);
    pool_kernel<<<waveNodeBlocks, 256, 0, stream>>>(bufB, bat, sums, cnts, N);
    final_kernel<<<(G * 32 + 255) / 256, 256, 0, stream>>>(sums, cnts, Wfc, bfc, out, G);
}